// Pew_LSTM_19619410608433
// MI455X (gfx1250) — compile-verified
//
#include <hip/hip_runtime.h>
#include <hip/hip_bf16.h>
#include <stdint.h>

// ---------------- problem constants (from reference) ----------------
#define SEQ_T   1024                  // timesteps
#define NCH     5                     // 4 weather channels + 1 input channel
#define BLK     128                   // threads/block = batch rows per TDM tile (4 waves)
#define TT      8                     // timesteps per tile
#define NT      (SEQ_T / TT)          // 128 tiles
#define ROW_DW  (TT * NCH)            // 40 data dwords per row
#define ROW_PAD (ROW_DW + ROW_DW / 8) // 45 dwords: TDM inserts 1 pad dword per 8 -> odd
                                      // stride -> conflict-free LDS banks across 32 lanes

#if defined(__AMDGCN__) &&                                   \
    __has_builtin(__builtin_amdgcn_tensor_load_to_lds) &&    \
    __has_builtin(__builtin_amdgcn_tensor_store_from_lds) && \
    __has_builtin(__builtin_amdgcn_s_wait_tensorcnt)
#define USE_TDM 1
#else
#define USE_TDM 0
#endif

typedef unsigned int v4u __attribute__((ext_vector_type(4)));
typedef int          v8i __attribute__((ext_vector_type(8)));
typedef int          v4i __attribute__((ext_vector_type(4)));

// sigmoid(x) = rcp(1 + exp2(-x*log2e))      (saturates correctly at +-inf)
__device__ __forceinline__ float fsigmoid(float x) {
  float e = __builtin_amdgcn_exp2f(x * -1.442695040888963f);
  return __builtin_amdgcn_rcpf(1.0f + e);
}
// tanh(x) = 1 - 2*rcp(exp2(2x*log2e) + 1)   (-> +1 / -1 at +-inf, no inf/inf)
__device__ __forceinline__ float ftanh(float x) {
  float e = __builtin_amdgcn_exp2f(x * 2.885390081777927f);
  return 1.0f - 2.0f * __builtin_amdgcn_rcpf(e + 1.0f);
}

#if USE_TDM
// Tensor DMA descriptor per CDNA5 ISA (2D tile, 4-byte elems).
// tensor_dim0 = tensor_dim1 = 1<<20 (tiles are always physically in-bounds -> no clip).
__device__ __forceinline__ void tdm_groups(v4u& g0, v8i& g1,
    uint64_t gaddr, uint32_t lds, uint32_t flags,
    uint32_t tile0, uint32_t tile1, uint32_t stride0) {
  g0[0] = 1u;                                                     // count=1 (valid)
  g0[1] = lds;                                                    // lds_addr (bytes)
  g0[2] = (uint32_t)gaddr;                                        // global_addr[31:0]
  g0[3] = ((uint32_t)(gaddr >> 32) & 0x01FFFFFFu) | (2u << 30);   // addr[56:32], type=2
  g1[0] = (int)(flags | (2u << 16));       // data_size=4B | pad flags
  g1[1] = 0;                               // no atomic barrier; tensor_dim0 lo16=0
  g1[2] = 0x0010;                          // tensor_dim0 hi16 (1<<20); dim1 lo16=0
  g1[3] = (int)(0x0010u | (tile0 << 16));  // tensor_dim1 hi16; tile_dim0
  g1[4] = (int)tile1;                      // tile_dim1; tile_dim2=0 (2D)
  g1[5] = (int)stride0;                    // tensor_dim0_stride lo32 (elements)
  g1[6] = 0;                               // stride0 hi16 = 0; dim1_stride lo16 = 0
  g1[7] = 0;
}
// This toolchain's TDM builtins take 6 args:
// (v4u g0, v8i g1, v4i g2, v4i g3, v8i extra, i32 cpol). 2D tensor -> g2/g3/extra = 0.
__device__ __forceinline__ void tdm_load_tile(uint64_t gaddr, uint32_t lds,
    uint32_t tile0, uint32_t tile1, uint32_t stride0, uint32_t flags) {
  v4u g0; v8i g1;
  v4i gz4 = {0, 0, 0, 0};
  v8i gz8 = {0, 0, 0, 0, 0, 0, 0, 0};
  tdm_groups(g0, g1, gaddr, lds, flags, tile0, tile1, stride0);
  __builtin_amdgcn_tensor_load_to_lds(g0, g1, gz4, gz4, gz8, 0);
}
__device__ __forceinline__ void tdm_store_tile(uint64_t gaddr, uint32_t lds,
    uint32_t tile0, uint32_t tile1, uint32_t stride0) {
  v4u g0; v8i g1;
  v4i gz4 = {0, 0, 0, 0};
  v8i gz8 = {0, 0, 0, 0, 0, 0, 0, 0};
  tdm_groups(g0, g1, gaddr, lds, 0u, tile0, tile1, stride0);
  __builtin_amdgcn_tensor_store_from_lds(g0, g1, gz4, gz4, gz8, 0);
}
#endif

__global__ __launch_bounds__(BLK) void pew_lstm_kernel(
    const float* __restrict__ inp,
    const float* __restrict__ W1, const float* __restrict__ U1,
    const float* __restrict__ V1, const float* __restrict__ B1,
    const float* __restrict__ W2, const float* __restrict__ U2,
    const float* __restrict__ V2, const float* __restrict__ B2,
    const float* __restrict__ FCW, const float* __restrict__ FCB,
    float* __restrict__ out)
{
  const int tid = threadIdx.x;
  const int b0  = blockIdx.x * BLK;

  // H=1 => all weights are tiny; keep every coefficient in registers.
  float w1[4], u1[4], v1[16], bb1[4], w2[4], u2[4], v2[16], bb2[4];
#pragma unroll
  for (int j = 0; j < 4; ++j) {
    w1[j] = W1[j]; u1[j] = U1[j]; bb1[j] = B1[j];
    w2[j] = W2[j]; u2[j] = U2[j]; bb2[j] = B2[j];
  }
#pragma unroll
  for (int j = 0; j < 16; ++j) { v1[j] = V1[j]; v2[j] = V2[j]; }
  const float fcw = FCW[0], fcb0 = FCB[0];

  float h1 = 0.f, c1 = 0.f, h2 = 0.f, c2 = 0.f;

  // One fused two-layer LSTM step. Weather partials (state-independent) are
  // tree-summed off the critical path; only h*U / h*W sit on the serial chain.
  auto step = [&](float wA, float wB, float wC, float wD, float xT) -> float {
    float p1[4], p2[4];
#pragma unroll
    for (int j = 0; j < 4; ++j) {
      p1[j] = (fmaf(wA, v1[j], wB * v1[4 + j]) + fmaf(wC, v1[8 + j], wD * v1[12 + j]))
            + fmaf(xT, w1[j], bb1[j]);
      p2[j] = (fmaf(wA, v2[j], wB * v2[4 + j]) + fmaf(wC, v2[8 + j], wD * v2[12 + j]))
            + bb2[j];
    }
    const float i1 = fsigmoid(fmaf(h1, u1[0], p1[0]));
    const float f1 = fsigmoid(fmaf(h1, u1[1], p1[1]));
    const float n1 = ftanh   (fmaf(h1, u1[2], p1[2]));
    const float o1 = fsigmoid(fmaf(h1, u1[3], p1[3]));
    c1 = fmaf(f1, c1, i1 * n1);
    h1 = o1 * ftanh(c1);
    const float i2 = fsigmoid(fmaf(h1, w2[0], fmaf(h2, u2[0], p2[0])));
    const float f2 = fsigmoid(fmaf(h1, w2[1], fmaf(h2, u2[1], p2[1])));
    const float n2 = ftanh   (fmaf(h1, w2[2], fmaf(h2, u2[2], p2[2])));
    const float o2 = fsigmoid(fmaf(h1, w2[3], fmaf(h2, u2[3], p2[3])));
    c2 = fmaf(f2, c2, i2 * n2);
    h2 = o2 * ftanh(c2);
    return fmaf(h2, fcw, fcb0);
  };

#if USE_TDM
  // Double-buffered TDM pipeline: tile = 128 batch rows x (8 steps * 5 ch) floats.
  __shared__ float s_in [2][BLK * ROW_PAD];   // 2 * 23040 B (TDM-padded rows)
  __shared__ float s_out[2][BLK * TT];        // 2 *  4096 B output staging
  const uint32_t lin0  = (uint32_t)(uintptr_t)&s_in [0][0];
  const uint32_t lin1  = (uint32_t)(uintptr_t)&s_in [1][0];
  const uint32_t lout0 = (uint32_t)(uintptr_t)&s_out[0][0];
  const uint32_t lout1 = (uint32_t)(uintptr_t)&s_out[1][0];
  const uint64_t in_base  = (uint64_t)(uintptr_t)inp + (uint64_t)b0 * (SEQ_T * NCH) * 4u;
  const uint64_t out_base = (uint64_t)(uintptr_t)out + (uint64_t)b0 * SEQ_T * 4u;
  const uint32_t PADF = (1u << 20) | (2u << 22);  // pad_enable, interval=8dw, amount=1dw

  // TDM ignores EXEC and TENSORcnt is per-wave: only wave 0 issues & waits.
  if (tid == 0)
    tdm_load_tile(in_base, lin0, ROW_DW, BLK, SEQ_T * NCH, PADF);

  int buf = 0;
  for (int tile = 0; tile < NT; ++tile) {
    if (tid == 0) {
      if (tile + 1 < NT) {
        tdm_load_tile(in_base + (uint64_t)(tile + 1) * (ROW_DW * 4),
                      (buf ? lin0 : lin1), ROW_DW, BLK, SEQ_T * NCH, PADF);
        // TDM ops complete in issue order: waiting for load(tile) also retires
        // store(tile-2), making its staging buffer reusable below.
        if (tile == 0) __builtin_amdgcn_s_wait_tensorcnt(1);
        else           __builtin_amdgcn_s_wait_tensorcnt(2);
      } else {
        __builtin_amdgcn_s_wait_tensorcnt(1);
      }
    }
    __syncthreads();                 // tile data visible to all 4 waves

    const float* row  = &s_in [buf][tid * ROW_PAD];
    float*       orow = &s_out[buf][tid * TT];
#pragma unroll
    for (int t = 0; t < TT; ++t) {
      const int i0 = t * NCH;        // padded LDS index: idx + idx/8 (compile-time)
      const float wA = row[i0 + 0 + ((i0 + 0) >> 3)];
      const float wB = row[i0 + 1 + ((i0 + 1) >> 3)];
      const float wC = row[i0 + 2 + ((i0 + 2) >> 3)];
      const float wD = row[i0 + 3 + ((i0 + 3) >> 3)];
      const float xT = row[i0 + 4 + ((i0 + 4) >> 3)];
      orow[t] = step(wA, wB, wC, wD, xT);
    }
    __syncthreads();                 // all reads of s_in[buf] + writes of s_out[buf] done

    if (tid == 0)                    // coalesced 2D store: rows=batch, stride=SEQ_T
      tdm_store_tile(out_base + (uint64_t)tile * (TT * 4),
                     (buf ? lout1 : lout0), TT, BLK, SEQ_T);
    buf ^= 1;
  }
  if (tid == 0) __builtin_amdgcn_s_wait_tensorcnt(0);
#else
  // Fallback (toolchain without TDM builtins): direct per-thread streaming.
  const float* __restrict__ mi = inp + (size_t)(b0 + tid) * (SEQ_T * NCH);
  float*       __restrict__ mo = out + (size_t)(b0 + tid) * SEQ_T;
  for (int t = 0; t < SEQ_T; ++t)
    mo[t] = step(mi[t * NCH + 0], mi[t * NCH + 1], mi[t * NCH + 2],
                 mi[t * NCH + 3], mi[t * NCH + 4]);
#endif
}

extern "C" void kernel_launch(void* const* d_in, const int* in_sizes, int n_in,
                              void* d_out, int out_size, void* d_ws, size_t ws_size,
                              hipStream_t stream) {
  (void)n_in; (void)out_size; (void)d_ws; (void)ws_size;
  const int B = in_sizes[0] / (SEQ_T * NCH);   // 8192
  pew_lstm_kernel<<<B / BLK, BLK, 0, stream>>>(
      (const float*)d_in[0],  // input (B, T, 5)
      (const float*)d_in[1], (const float*)d_in[2], (const float*)d_in[3],
      (const float*)d_in[4],  // W1 U1 V1 b1
      (const float*)d_in[5], (const float*)d_in[6], (const float*)d_in[7],
      (const float*)d_in[8],  // W2 U2 V2 b2
      (const float*)d_in[9], (const float*)d_in[10],  // fcW fcb
      (float*)d_out);
}